// AdaptiveMessagePassingLayer_60318520705632
// MI455X (gfx1250) — compile-verified
//
#include <hip/hip_runtime.h>

typedef __attribute__((ext_vector_type(2))) float v2f;
typedef __attribute__((ext_vector_type(8))) float v8f;

#define D_IN   128
#define D_OUT  128
#define NREL   16
#define LDS_STRIDE 132          // 128 + 4 floats pad -> conflict-free ds_load_b64
#define BLOCK_ROWS 64           // rows staged in LDS per iteration
#define ROW_ITERS  4            // 256 rows per block
#define N_NODES    524288
#define BUF_FLOATS (BLOCK_ROWS * LDS_STRIDE)   // 8448 floats = 33792 B / buffer

// ---------------------------------------------------------------------------
// Kernel 1: W_eff = sum_r s_r * W_r   (16 x 128 x 128 -> 128 x 128)
// ---------------------------------------------------------------------------
__global__ __launch_bounds__(256) void fold_weights_kernel(
    const float4* __restrict__ W,   // [16][4096] float4 view of [16,128,128]
    const float*  __restrict__ s,   // [16]
    float4*       __restrict__ weff)
{
    const int i = blockIdx.x * 256 + threadIdx.x;   // 0..4095
    float4 acc; acc.x = 0.f; acc.y = 0.f; acc.z = 0.f; acc.w = 0.f;
#pragma unroll
    for (int r = 0; r < NREL; ++r) {
        const float  sc = s[r];
        const float4 w  = W[r * 4096 + i];
        acc.x += sc * w.x;
        acc.y += sc * w.y;
        acc.z += sc * w.z;
        acc.w += sc * w.w;
    }
    weff[i] = acc;
}

// ---------------------------------------------------------------------------
// Async global->LDS copy, 16B per lane (GLOBAL_LOAD_ASYNC_TO_LDS_B128,
// tracked with ASYNCcnt; no VGPR data round-trip).
// ---------------------------------------------------------------------------
__device__ __forceinline__ void async_b128_to_lds(unsigned lds_byte_off,
                                                  const float* gsrc) {
    asm volatile("global_load_async_to_lds_b128 %0, %1, off"
                 :
                 : "v"(lds_byte_off), "v"(gsrc)
                 : "memory");
}

__device__ __forceinline__ void wait_async0() {
    asm volatile("s_wait_asynccnt 0x0" ::: "memory");
}

// ---------------------------------------------------------------------------
// Kernel 2: Y[N,128] = X[N,128] @ Weff[128,128] via V_WMMA_F32_16X16X4_F32.
// Block = 128 threads (4 waves). Wave w owns output columns [32w, 32w+32).
// B fragments (full K) in registers; X double-buffered in LDS via async DMA.
// ---------------------------------------------------------------------------
__global__ __launch_bounds__(128) void rgcn_gemm_kernel(
    const float* __restrict__ X,
    const float* __restrict__ Wf,   // [128][128] row-major (K, N)
    float*       __restrict__ Y)
{
    __shared__ float sX[2][BUF_FLOATS];   // 2 x 33792 B

    const int tid  = threadIdx.x;
    const int lane = tid & 31;
    const int wave = tid >> 5;        // 0..3
    const int half = lane >> 4;       // 0 or 1
    const int m16  = lane & 15;

    // LDS byte offset of sX[0] within the workgroup's LDS allocation
    // (generic pointer low 32 bits == LDS address per ISA flat->LDS rule).
    const unsigned sbase = (unsigned)(uintptr_t)(&sX[0][0]);

    // ---- Load B fragments for this wave's 32 columns, all 32 k-chunks ----
    // B (4x16) layout: lane L holds N = L%16; VGPR v holds K = 4kc + v + 2*(L/16)
    v2f b[2][32];
#pragma unroll
    for (int t = 0; t < 2; ++t) {
        const int col = wave * 32 + t * 16 + m16;
#pragma unroll
        for (int kc = 0; kc < 32; ++kc) {
            const int k0 = kc * 4 + half * 2;
            v2f f;
            f.x = Wf[k0 * D_OUT + col];
            f.y = Wf[(k0 + 1) * D_OUT + col];
            b[t][kc] = f;
        }
    }

    const long rowbase = (long)blockIdx.x * (BLOCK_ROWS * ROW_ITERS);

    // ---- stage helper: 64 rows (32 KB) -> LDS buffer, 16 async b128/thread ----
    // (written out inline below to keep everything in one function)
#define STAGE(bufsel, rbexpr)                                                   \
    do {                                                                        \
        const long _rb = (rbexpr);                                              \
        _Pragma("unroll")                                                       \
        for (int l = 0; l < 16; ++l) {                                          \
            const int q   = l * 128 + tid;   /* 0..2047 quad index */           \
            const int row = q >> 5;          /* 32 quads per row   */           \
            const int qc  = q & 31;                                             \
            const unsigned lds = sbase + (unsigned)((bufsel) * BUF_FLOATS +     \
                                 row * LDS_STRIDE + qc * 4) * 4u;               \
            async_b128_to_lds(lds, X + (_rb + row) * D_IN + qc * 4);            \
        }                                                                       \
    } while (0)

    // Prologue: fetch tile 0 into buffer 0.
    STAGE(0, rowbase);

    for (int it = 0; it < ROW_ITERS; ++it) {
        const int  cur = it & 1;
        const long rb  = rowbase + (long)it * BLOCK_ROWS;

        // All of this wave's async writes to buf[cur] done; barrier makes them
        // visible to all waves AND guarantees everyone finished reading
        // buf[1-cur] (those reads were in iteration it-1, before this point).
        wait_async0();
        __syncthreads();

        // Prefetch next tile into the other buffer, overlapped with compute.
        if (it + 1 < ROW_ITERS)
            STAGE(1 - cur, rowbase + (long)(it + 1) * BLOCK_ROWS);

        // ---- compute 4 row-tiles of 16 rows each from buf[cur] ----
#pragma unroll
        for (int rt = 0; rt < 4; ++rt) {
            v8f c0 = {};
            v8f c1 = {};
            // A (16x4) layout: lane L holds M = L%16; VGPR v: K = 4kc + v + 2*(L/16)
            const float* arow = &sX[cur][(rt * 16 + m16) * LDS_STRIDE + half * 2];
#pragma unroll
            for (int kc = 0; kc < 32; ++kc) {
                const v2f a = *(const v2f*)(arow + kc * 4);
                c0 = __builtin_amdgcn_wmma_f32_16x16x4_f32(
                        false, a, false, b[0][kc], (short)0, c0, false, false);
                c1 = __builtin_amdgcn_wmma_f32_16x16x4_f32(
                        false, a, false, b[1][kc], (short)0, c1, false, false);
            }
            // C/D layout: VGPR j -> row (j + 8*half), col = N = m16
            const long orow = rb + rt * 16 + half * 8;
#pragma unroll
            for (int j = 0; j < 8; ++j) {
                Y[(orow + j) * D_OUT + wave * 32 + m16]      = c0[j];
                Y[(orow + j) * D_OUT + wave * 32 + 16 + m16] = c1[j];
            }
        }
    }
#undef STAGE
}

// ---------------------------------------------------------------------------
extern "C" void kernel_launch(void* const* d_in, const int* in_sizes, int n_in,
                              void* d_out, int out_size, void* d_ws, size_t ws_size,
                              hipStream_t stream) {
    const float* X  = (const float*)d_in[0];   // [524288, 128]
    const float* W  = (const float*)d_in[1];   // [16, 128, 128]
    const float* s  = (const float*)d_in[2];   // [16, 1]
    float*       Y  = (float*)d_out;           // [524288, 128]
    float*       wf = (float*)d_ws;            // 64 KB scratch for W_eff

    fold_weights_kernel<<<16, 256, 0, stream>>>(
        (const float4*)W, s, (float4*)wf);

    const int rows_per_block = BLOCK_ROWS * ROW_ITERS;          // 256
    const int nblocks = N_NODES / rows_per_block;               // 2048
    rgcn_gemm_kernel<<<nblocks, 128, 0, stream>>>(X, wf, Y);
}